// GPTBigCodeAttention_53266184405358
// MI455X (gfx1250) — compile-verified
//
#include <hip/hip_runtime.h>

typedef float v2f __attribute__((ext_vector_type(2)));
typedef float v8f __attribute__((ext_vector_type(8)));

#define EMBED    4096
#define NHEADS   32
#define HDIM     128
#define BATCH    64
#define KVLEN    4096
#define QKV_N    (EMBED + 2 * HDIM)   /* 4352 */
#define PAST_ROW (2 * HDIM)           /* 256  */
#define MASK_VALUE (-3.4028234663852886e38f)

static __device__ __forceinline__ v8f wmma4(v2f a, v2f b, v8f c) {
  // D = A(16x4) * B(4x16) + C, fp32 throughout.
  return __builtin_amdgcn_wmma_f32_16x16x4_f32(false, a, false, b, (short)0, c,
                                               false, false);
}

// ---------------------------------------------------------------------------
// Row-major GEMM + bias: D[M x N] = A[M x K] * B[K x N] + bias[N].
// One wave owns a (16*MT) x 16 strip of D: each B fragment is loaded ONCE
// and reused across MT accumulators, so the large B operand is streamed from
// HBM exactly once. grid = (N/16, M/(16*MT)), block = 32 (one wave).
// ---------------------------------------------------------------------------
template <int MT>
__global__ void gemm_bias_kernel(const float* __restrict__ A,
                                 const float* __restrict__ B,
                                 const float* __restrict__ bias,
                                 float* __restrict__ D,
                                 int K, int lda, int ldb, int ldd) {
  const int tn   = blockIdx.x * 16;
  const int tm0  = blockIdx.y * 16 * MT;
  const int lane = threadIdx.x & 31;
  const int c    = lane & 15;   // col within tile (and A row)
  const int kh   = lane >> 4;   // k-half select

  v8f acc[MT];
#pragma unroll
  for (int mt = 0; mt < MT; ++mt) acc[mt] = (v8f){0.f,0.f,0.f,0.f,0.f,0.f,0.f,0.f};

  const float* arow[MT];
#pragma unroll
  for (int mt = 0; mt < MT; ++mt)
    arow[mt] = A + (long)(tm0 + mt * 16 + c) * lda + 2 * kh;
  const float* bcol = B + tn + c + (long)(2 * kh) * ldb;

  for (int k0 = 0; k0 < K; k0 += 16) {
#pragma unroll
    for (int u = 0; u < 4; ++u) {
      const int k = k0 + 4 * u;
      v2f bfrag;
      bfrag.x = bcol[(long)k * ldb];
      bfrag.y = bcol[(long)(k + 1) * ldb];
#pragma unroll
      for (int mt = 0; mt < MT; ++mt) {
        v2f a;
        a.x = arow[mt][k];
        a.y = arow[mt][k + 1];
        acc[mt] = wmma4(a, bfrag, acc[mt]);
      }
    }
  }

  const float bv = bias ? bias[tn + c] : 0.0f;
#pragma unroll
  for (int mt = 0; mt < MT; ++mt) {
#pragma unroll
    for (int j = 0; j < 8; ++j) {
      const int m = tm0 + mt * 16 + j + 8 * kh;
      D[(long)m * ldd + tn + c] = acc[mt][j] + bv;
    }
  }
}

// ---------------------------------------------------------------------------
// layer_past functional update: copy (B,KVLEN,256) and splice the new
// key/value row (qkv[:, 4096:4352]) at row key_length-1. float4 streaming.
// ---------------------------------------------------------------------------
__global__ void past_copy_kernel(const float* __restrict__ past_in,
                                 const float* __restrict__ qkv,
                                 const int* __restrict__ key_length,
                                 float* __restrict__ past_out) {
  const long total4 = (long)BATCH * KVLEN * PAST_ROW / 4;
  const int  upd    = *key_length - 1;
  const long stride = (long)gridDim.x * blockDim.x;
  const float4* src = (const float4*)past_in;
  float4*       dst = (float4*)past_out;

  for (long i = (long)blockIdx.x * blockDim.x + threadIdx.x; i < total4;
       i += stride) {
    const long fidx = i * 4;
    const int  cc   = (int)(fidx % PAST_ROW);
    const long rowb = fidx / PAST_ROW;
    const int  r    = (int)(rowb % KVLEN);
    const int  b    = (int)(rowb / KVLEN);
    float4 v;
    if (r == upd) {
      const float* s = qkv + (long)b * QKV_N + EMBED + cc;
      v = make_float4(s[0], s[1], s[2], s[3]);
    } else {
      v = src[i];
    }
    dst[i] = v;
  }
}

// ---------------------------------------------------------------------------
// scores[b,h,p] = mask ? (Q[b,h,:] . K[b,p,:]) * scale : MASK_VALUE
// WMMA: A = Q (32x128, lda=128), B[k][p] = K[p][k]. One wave handles BOTH
// head tiles (MT=2) so each K row is streamed once.
// grid = (KVLEN/16, 1, BATCH), block = 32.
// ---------------------------------------------------------------------------
__global__ void scores_kernel(const float* __restrict__ qkv,
                              const float* __restrict__ past,
                              const unsigned char* __restrict__ mask,
                              float* __restrict__ scores) {
  const int b    = blockIdx.z;
  const int tp   = blockIdx.x * 16;  // key-position tile
  const int lane = threadIdx.x & 31;
  const int c    = lane & 15;
  const int kh   = lane >> 4;

  const float* Q  = qkv + (long)b * QKV_N;                 // Q[h*128 + d]
  const float* Kb = past + (long)b * KVLEN * PAST_ROW;     // K[p*256 + d]

  v8f acc0 = {0.f,0.f,0.f,0.f,0.f,0.f,0.f,0.f};
  v8f acc1 = {0.f,0.f,0.f,0.f,0.f,0.f,0.f,0.f};
  const float* arow0 = Q + (long)(c)      * HDIM + 2 * kh;
  const float* arow1 = Q + (long)(16 + c) * HDIM + 2 * kh;
  const float* brow  = Kb + (long)(tp + c) * PAST_ROW + 2 * kh;

#pragma unroll
  for (int k0 = 0; k0 < HDIM; k0 += 4) {
    v2f bb;
    bb.x = brow[k0];       // B[k][p] = K[p][k]: contiguous k pair
    bb.y = brow[k0 + 1];
    v2f a0, a1;
    a0.x = arow0[k0]; a0.y = arow0[k0 + 1];
    a1.x = arow1[k0]; a1.y = arow1[k0 + 1];
    acc0 = wmma4(a0, bb, acc0);
    acc1 = wmma4(a1, bb, acc1);
  }

  const float scale = 0.08838834764831845f;  // 1/sqrt(128)
  const int   p     = tp + c;
  const bool  keep  = mask[(long)b * KVLEN + p] != 0;
  float* srow = scores + (long)b * NHEADS * KVLEN + p;
#pragma unroll
  for (int j = 0; j < 8; ++j) {
    const int h0 = j + 8 * kh;
    srow[(long)h0 * KVLEN]        = keep ? acc0[j] * scale : MASK_VALUE;
    srow[(long)(16 + h0) * KVLEN] = keep ? acc1[j] * scale : MASK_VALUE;
  }
}

// ---------------------------------------------------------------------------
// In-place row softmax over KVLEN. grid = BATCH*NHEADS rows, block = 256.
// ---------------------------------------------------------------------------
__global__ void softmax_kernel(float* __restrict__ scores) {
  float* s = scores + (long)blockIdx.x * KVLEN;
  __shared__ float red[256];
  const int tid = threadIdx.x;

  float m = -__builtin_inff();
  for (int i = tid; i < KVLEN; i += 256) m = fmaxf(m, s[i]);
  red[tid] = m;
  __syncthreads();
  for (int off = 128; off > 0; off >>= 1) {
    if (tid < off) red[tid] = fmaxf(red[tid], red[tid + off]);
    __syncthreads();
  }
  m = red[0];
  __syncthreads();

  float sum = 0.f;
  for (int i = tid; i < KVLEN; i += 256) {
    const float e = __expf(s[i] - m);
    s[i] = e;
    sum += e;
  }
  red[tid] = sum;
  __syncthreads();
  for (int off = 128; off > 0; off >>= 1) {
    if (tid < off) red[tid] += red[tid + off];
    __syncthreads();
  }
  const float inv = 1.0f / red[0];
  for (int i = tid; i < KVLEN; i += 256) s[i] *= inv;
}

// ---------------------------------------------------------------------------
// attn[b,h,d] = sum_p probs[b,h,p] * V[b,p,d]
// WMMA: A = probs (32x4096), B = V (4096x128, ldb=256) -> coalesced B loads.
// One wave handles BOTH head tiles (MT=2) so V is streamed once per n-tile.
// grid = (HDIM/16, 1, BATCH), block = 32.
// ---------------------------------------------------------------------------
__global__ void attnv_kernel(const float* __restrict__ probs,
                             const float* __restrict__ past,
                             float* __restrict__ attn) {
  const int b    = blockIdx.z;
  const int tn   = blockIdx.x * 16;  // head-dim tile
  const int lane = threadIdx.x & 31;
  const int c    = lane & 15;
  const int kh   = lane >> 4;

  const float* P  = probs + (long)b * NHEADS * KVLEN;
  const float* Vb = past + (long)b * KVLEN * PAST_ROW + HDIM;

  v8f acc0 = {0.f,0.f,0.f,0.f,0.f,0.f,0.f,0.f};
  v8f acc1 = {0.f,0.f,0.f,0.f,0.f,0.f,0.f,0.f};
  const float* arow0 = P + (long)(c)      * KVLEN + 2 * kh;
  const float* arow1 = P + (long)(16 + c) * KVLEN + 2 * kh;
  const float* bcol  = Vb + tn + c + (long)(2 * kh) * PAST_ROW;

  for (int k0 = 0; k0 < KVLEN; k0 += 16) {
#pragma unroll
    for (int u = 0; u < 4; ++u) {
      const int k = k0 + 4 * u;
      v2f bb;
      bb.x = bcol[(long)k * PAST_ROW];
      bb.y = bcol[(long)(k + 1) * PAST_ROW];
      v2f a0, a1;
      a0.x = arow0[k]; a0.y = arow0[k + 1];
      a1.x = arow1[k]; a1.y = arow1[k + 1];
      acc0 = wmma4(a0, bb, acc0);
      acc1 = wmma4(a1, bb, acc1);
    }
  }

  float* drow = attn + (long)b * EMBED + tn + c;
#pragma unroll
  for (int j = 0; j < 8; ++j) {
    const int h0 = j + 8 * kh;
    drow[(long)h0 * HDIM]        = acc0[j];
    drow[(long)(16 + h0) * HDIM] = acc1[j];
  }
}

// ---------------------------------------------------------------------------
extern "C" void kernel_launch(void* const* d_in, const int* in_sizes, int n_in,
                              void* d_out, int out_size, void* d_ws,
                              size_t ws_size, hipStream_t stream) {
  const float*         hidden     = (const float*)d_in[0];
  const float*         past       = (const float*)d_in[1];
  const unsigned char* mask       = (const unsigned char*)d_in[2];
  const float*         c_attn_w   = (const float*)d_in[3];
  const float*         c_attn_b   = (const float*)d_in[4];
  const float*         c_proj_w   = (const float*)d_in[5];
  const float*         c_proj_b   = (const float*)d_in[6];
  const int*           key_length = (const int*)d_in[7];

  float* out      = (float*)d_out;                    // (64, 4096)
  float* past_out = out + (long)BATCH * EMBED;        // (64, 4096, 256)

  float* qkv    = (float*)d_ws;                       // 64 x 4352
  float* scores = qkv + (long)BATCH * QKV_N;          // 64 x 32 x 4096
  float* attn   = scores + (long)BATCH * NHEADS * KVLEN;  // 64 x 4096

  // 1) qkv = hidden @ c_attn_w + c_attn_b   (wave owns full M=64 strip)
  gemm_bias_kernel<4><<<dim3(QKV_N / 16, 1), 32, 0, stream>>>(
      hidden, c_attn_w, c_attn_b, qkv, EMBED, EMBED, QKV_N, QKV_N);

  // 2) past_out = layer_past with row key_length-1 <- qkv[:, 4096:]
  past_copy_kernel<<<dim3(8192), 256, 0, stream>>>(past, qkv, key_length,
                                                   past_out);

  // 3) scores = scale * Q K^T, masked (K rows streamed once)
  scores_kernel<<<dim3(KVLEN / 16, 1, BATCH), 32, 0, stream>>>(
      qkv, past_out, mask, scores);

  // 4) softmax rows
  softmax_kernel<<<dim3(BATCH * NHEADS), 256, 0, stream>>>(scores);

  // 5) attn = probs @ V (V streamed once per n-tile)
  attnv_kernel<<<dim3(HDIM / 16, 1, BATCH), 32, 0, stream>>>(
      scores, past_out, attn);

  // 6) out = attn @ c_proj_w + c_proj_b  (wave owns full M=64 strip)
  gemm_bias_kernel<4><<<dim3(EMBED / 16, 1), 32, 0, stream>>>(
      attn, c_proj_w, c_proj_b, out, EMBED, EMBED, EMBED, EMBED);
}